// DGCNN_22084721836401
// MI455X (gfx1250) — compile-verified
//
#include <hip/hip_runtime.h>
#include <hip/hip_bf16.h>
#include <float.h>

typedef float v2f __attribute__((ext_vector_type(2)));
typedef float v8f __attribute__((ext_vector_type(8)));

#define D_TOT   97
#define KTOP    30
#define FSTRIDE 100   // feat row stride (even, >= 97)
#define PSTRIDE 104   // pooled row stride (multiple of 8, >= 97)
#define XPAD    24    // padded input feature count (multiple of 8, >= 20)
#define ZBUF_N  360   // zero scratch: >= max Kdim (352)

// ---------------------------------------------------------------------------
// fp32 WMMA GEMM: C[M x N] = act( A(MxK, lda) @ Bt(NxK, ldb)^T + bias )
// Bt is stored N-major with rows along K (pre-transposed, zero-padded so that
// Kdim is a multiple of 8 and every load below is in-bounds & 8B aligned).
// Out-of-range lanes read from a zero-filled scratch region instead of being
// masked per-load -> inner loop is pure global_load_b64 + v_wmma.
// One wave per 16x16 C tile; two independent accumulator chains.
// ---------------------------------------------------------------------------
__global__ void wmma_gemm_f32(const float* __restrict__ A,
                              const float* __restrict__ Bt,
                              const float* __restrict__ bias,
                              const float* __restrict__ zbuf,
                              float* __restrict__ C,
                              int M, int Kdim, int Ncols,
                              int lda, int ldb, int ldc, int mode) {
    const int lane = threadIdx.x;             // 0..31
    const int tilesN = (Ncols + 15) >> 4;
    const int tilesM = (M + 15) >> 4;
    const int tileIdx = blockIdx.x * blockDim.y + threadIdx.y;
    if (tileIdx >= tilesM * tilesN) return;   // uniform per wave (blockDim.x==32)
    const int tm = tileIdx / tilesN;
    const int tn = tileIdx % tilesN;

    const int mrow  = tm * 16 + (lane & 15);
    const int ncol  = tn * 16 + (lane & 15);
    const int khalf = (lane >> 4) * 2;        // 0 (lanes 0-15) or 2 (lanes 16-31)

    const bool aok = (mrow < M);
    const bool bok = (ncol < Ncols);
    // OOB lanes read zeros from zbuf -> no masking in the hot loop
    const float* ap = (aok ? (A  + (size_t)mrow * lda) : zbuf) + khalf;
    const float* bp = (bok ? (Bt + (size_t)ncol * ldb) : zbuf) + khalf;

    v8f acc0 = {0.f, 0.f, 0.f, 0.f, 0.f, 0.f, 0.f, 0.f};
    v8f acc1 = {0.f, 0.f, 0.f, 0.f, 0.f, 0.f, 0.f, 0.f};

    for (int kk = 0; kk < Kdim; kk += 8) {
        v2f a0 = *(const v2f*)(ap + kk);
        v2f b0 = *(const v2f*)(bp + kk);
        v2f a1 = *(const v2f*)(ap + kk + 4);
        v2f b1 = *(const v2f*)(bp + kk + 4);
        acc0 = __builtin_amdgcn_wmma_f32_16x16x4_f32(false, a0, false, b0,
                                                     (short)0, acc0, false, false);
        acc1 = __builtin_amdgcn_wmma_f32_16x16x4_f32(false, a1, false, b1,
                                                     (short)0, acc1, false, false);
    }
    v8f acc = acc0 + acc1;

    // D layout: VGPR v, lanes 0-15 -> row v ; lanes 16-31 -> row v+8 ; col = lane%16
    if (bok) {
        const float bv = (mode == 1) ? bias[ncol] : 0.f;
        const int rbase = tm * 16 + ((lane >> 4) << 3);
#pragma unroll
        for (int v = 0; v < 8; ++v) {
            const int r = rbase + v;
            if (r < M) {
                float val = acc[v] + bv;
                if (mode == 1) val = fmaxf(val, 0.f);
                C[(size_t)r * ldc + ncol] = val;
            }
        }
    }
}

// ---------------------------------------------------------------------------
// Weight prep: transpose (KxN row-major) -> (N x Kpad), zero-padded along K.
// ---------------------------------------------------------------------------
__global__ void transpose_pad(const float* __restrict__ in, float* __restrict__ out,
                              int K, int N, int Kpad) {
    int t = blockIdx.x * blockDim.x + threadIdx.x;
    if (t >= N * Kpad) return;
    int n = t / Kpad, k = t - n * Kpad;
    out[t] = (k < K) ? in[(size_t)k * N + n] : 0.f;
}

// Pad rows of an already N-major (rows x K) matrix to stride ldo with zeros.
__global__ void pad_rows(const float* __restrict__ in, float* __restrict__ out,
                         int rows, int K, int ldi, int ldo) {
    int t = blockIdx.x * blockDim.x + threadIdx.x;
    if (t >= rows * ldo) return;
    int r = t / ldo, k = t - r * ldo;
    out[t] = (k < K) ? in[(size_t)r * ldi + k] : 0.f;
}

// ---------------------------------------------------------------------------
// Graph-norm helpers
// ---------------------------------------------------------------------------
__global__ void fill_f32(float* p, float v, int n) {
    int i = blockIdx.x * blockDim.x + threadIdx.x;
    if (i < n) p[i] = v;
}

__global__ void deg_scatter(const int* __restrict__ dst, float* __restrict__ deg, int E) {
    int e = blockIdx.x * blockDim.x + threadIdx.x;
    if (e < E) atomicAdd(&deg[dst[e]], 1.0f);
}

__global__ void deg_to_dis(float* d, int n) {
    int i = blockIdx.x * blockDim.x + threadIdx.x;
    if (i < n) d[i] = rsqrtf(fmaxf(d[i], 1.0f));
}

// agg[i][j] = b[j] + hw[i][j] * dis[i]^2   (bias + self-loop term), dout = 1<<shift
__global__ void gcn_init(const float* __restrict__ hw, const float* __restrict__ dis,
                         const float* __restrict__ bias, float* __restrict__ agg,
                         int n, int shift) {
    int t = blockIdx.x * blockDim.x + threadIdx.x;
    if (t >= (n << shift)) return;
    int i = t >> shift, j = t & ((1 << shift) - 1);
    float di = dis[i];
    agg[t] = bias[j] + hw[t] * di * di;
}

// agg[dst][j] += hw[src][j] * dis[src]*dis[dst]
__global__ void gcn_scatter(const float* __restrict__ hw, const float* __restrict__ dis,
                            const int* __restrict__ src, const int* __restrict__ dst,
                            float* __restrict__ agg, int E, int shift) {
    long long t = (long long)blockIdx.x * blockDim.x + threadIdx.x;
    if (t >= ((long long)E << shift)) return;
    int e = (int)(t >> shift), j = (int)(t & ((1 << shift) - 1));
    int s = src[e], d = dst[e];
    float w = dis[s] * dis[d];
    atomicAdd(&agg[((size_t)d << shift) + j], hw[((size_t)s << shift) + j] * w);
}

// feat[i][colOff + j] = tanh(agg[i][j])
__global__ void gcn_tanh(const float* __restrict__ agg, float* __restrict__ feat,
                         int n, int shift, int colOff) {
    int t = blockIdx.x * blockDim.x + threadIdx.x;
    if (t >= (n << shift)) return;
    int i = t >> shift, j = t & ((1 << shift) - 1);
    feat[(size_t)i * FSTRIDE + colOff + j] = tanhf(agg[t]);
}

// ---------------------------------------------------------------------------
// SortPooling: one block per graph; batch[] is sorted -> contiguous range via
// binary search. Replicates lexsort((-last, batch)): value desc, index asc.
// ---------------------------------------------------------------------------
__global__ void sortpool_kernel(const float* __restrict__ feat,
                                const int* __restrict__ batch,
                                float* __restrict__ pooled, int n) {
    const int g = blockIdx.x;
    const int t = threadIdx.x;
    __shared__ float sV[128];
    __shared__ int   sI[128];
    __shared__ float lastVal;
    __shared__ int   lastIdx;

    int lo = 0, hi = n;
    while (lo < hi) { int m = (lo + hi) >> 1; if (batch[m] < g) lo = m + 1; else hi = m; }
    const int lb = lo;
    hi = n;
    while (lo < hi) { int m = (lo + hi) >> 1; if (batch[m] < g + 1) lo = m + 1; else hi = m; }
    const int ub = lo;

    if (t == 0) { lastVal = FLT_MAX; lastIdx = -1; }
    __syncthreads();

    for (int k = 0; k < KTOP; ++k) {
        const float lv = lastVal;
        const int   li = lastIdx;
        float bestV = -FLT_MAX;
        int   bestI = 0x7fffffff;
        for (int i = lb + t; i < ub; i += blockDim.x) {
            float v = feat[(size_t)i * FSTRIDE + (D_TOT - 1)];
            if (v < lv || (v == lv && i > li)) {             // strictly after (lv,li)
                if (v > bestV || (v == bestV && i < bestI)) { bestV = v; bestI = i; }
            }
        }
        sV[t] = bestV; sI[t] = bestI;
        __syncthreads();
        for (int s = blockDim.x >> 1; s > 0; s >>= 1) {
            if (t < s) {
                if (sV[t + s] > sV[t] || (sV[t + s] == sV[t] && sI[t + s] < sI[t])) {
                    sV[t] = sV[t + s]; sI[t] = sI[t + s];
                }
            }
            __syncthreads();
        }
        const int   node  = sI[0];
        const float nval  = sV[0];
        const bool  valid = (k < ub - lb) && (node != 0x7fffffff);
        float* dstRow = pooled + ((size_t)g * KTOP + k) * PSTRIDE;
        for (int j = t; j < PSTRIDE; j += blockDim.x)
            dstRow[j] = (valid && j < D_TOT) ? feat[(size_t)node * FSTRIDE + j] : 0.f;
        __syncthreads();
        if (t == 0) { lastVal = nval; lastIdx = node; }
        __syncthreads();
    }
}

// ---------------------------------------------------------------------------
// CNN head helpers
// ---------------------------------------------------------------------------
__global__ void maxpool_kernel(const float* __restrict__ y1, float* __restrict__ y1p, int G) {
    int t = blockIdx.x * blockDim.x + threadIdx.x;
    int total = G * 16 * 15;
    if (t >= total) return;
    int g = t / (16 * 15);
    int r = t - g * 16 * 15;
    int c = r / 15;
    int tt = r - c * 15;
    float a = y1[((size_t)g * 30 + 2 * tt) * 16 + c];
    float b = y1[((size_t)g * 30 + 2 * tt + 1) * 16 + c];
    y1p[t] = fmaxf(a, b);
}

__global__ void conv2_kernel(const float* __restrict__ y1p, const float* __restrict__ w,
                             const float* __restrict__ b, float* __restrict__ y2, int G) {
    int t = blockIdx.x * blockDim.x + threadIdx.x;
    int total = G * 32 * 11;
    if (t >= total) return;
    int g = t / (32 * 11);
    int r = t - g * 32 * 11;
    int o = r / 11;
    int tt = r - o * 11;
    float acc = b[o];
#pragma unroll
    for (int c = 0; c < 16; ++c) {
        const float* xrow = y1p + ((size_t)g * 16 + c) * 15 + tt;
        const float* wrow = w + ((size_t)o * 16 + c) * 5;
#pragma unroll
        for (int j = 0; j < 5; ++j) acc += xrow[j] * wrow[j];
    }
    y2[(size_t)g * 352 + o * 11 + tt] = fmaxf(acc, 0.f);
}

__global__ void mlp2_kernel(const float* __restrict__ a, const float* __restrict__ w,
                            const float* __restrict__ b, float* __restrict__ out, int G) {
    int g = blockIdx.x * blockDim.x + threadIdx.x;
    if (g >= G) return;
    float acc = b[0];
#pragma unroll
    for (int j = 0; j < 32; ++j) acc += a[(size_t)g * 32 + j] * w[j];
    out[g] = acc;
}

// ---------------------------------------------------------------------------
// Host-side orchestration
// ---------------------------------------------------------------------------
static inline int cdiv(long long a, long long b) { return (int)((a + b - 1) / b); }

static void launch_gemm(const float* A, int lda, const float* Bt, int ldb,
                        const float* bias, const float* zbuf, float* C, int ldc,
                        int M, int K, int N, int mode, hipStream_t stream) {
    int tiles = cdiv(M, 16) * cdiv(N, 16);
    dim3 blk(32, 4);
    dim3 grd(cdiv(tiles, 4));
    wmma_gemm_f32<<<grd, blk, 0, stream>>>(A, Bt, bias, zbuf, C, M, K, N, lda, ldb, ldc, mode);
}

extern "C" void kernel_launch(void* const* d_in, const int* in_sizes, int n_in,
                              void* d_out, int out_size, void* d_ws, size_t ws_size,
                              hipStream_t stream) {
    const float* x       = (const float*)d_in[0];
    const int*   edge    = (const int*)  d_in[1];
    const int*   batch   = (const int*)  d_in[2];
    const float* W[4]    = { (const float*)d_in[3], (const float*)d_in[5],
                             (const float*)d_in[7], (const float*)d_in[9] };
    const float* bW[4]   = { (const float*)d_in[4], (const float*)d_in[6],
                             (const float*)d_in[8], (const float*)d_in[10] };
    const float* conv1_w = (const float*)d_in[11];
    const float* conv1_b = (const float*)d_in[12];
    const float* conv2_w = (const float*)d_in[13];
    const float* conv2_b = (const float*)d_in[14];
    const float* mlp_w1  = (const float*)d_in[15];
    const float* mlp_b1  = (const float*)d_in[16];
    const float* mlp_w2  = (const float*)d_in[17];
    const float* mlp_b2  = (const float*)d_in[18];

    const int N = in_sizes[2];
    const int F = in_sizes[0] / N;   // 20
    const int E = in_sizes[1] / 2;
    const int H = 32;
    const int G = out_size;

    const int* src = edge;
    const int* dst = edge + E;

    // workspace carve-up (fp32, all offsets even -> 8B alignment holds)
    float* ws = (float*)d_ws;
    size_t o = 0;
    float* dis    = ws + o; o += (size_t)N;  o += (o & 1);
    float* zbuf   = ws + o; o += (size_t)ZBUF_N;
    float* xpad   = ws + o; o += (size_t)N * XPAD;
    float* hw     = ws + o; o += (size_t)N * H;
    float* agg    = ws + o; o += (size_t)N * H;
    float* feat   = ws + o; o += (size_t)N * FSTRIDE;
    float* pooled = ws + o; o += (size_t)G * KTOP * PSTRIDE;
    float* y1     = ws + o; o += (size_t)G * KTOP * 16;
    float* y1p    = ws + o; o += (size_t)G * 16 * 15;
    float* y2     = ws + o; o += (size_t)G * 352;
    float* m1     = ws + o; o += (size_t)G * 32;
    float* W0t    = ws + o; o += (size_t)H * XPAD;     // 32 x 24
    float* W1t    = ws + o; o += (size_t)H * H;        // 32 x 32
    float* W2t    = ws + o; o += (size_t)H * H;        // 32 x 32
    float* W3t    = ws + o; o += (size_t)1 * H;        // 1 x 32
    float* C1t    = ws + o; o += (size_t)16 * PSTRIDE; // 16 x 104
    float* M1t    = ws + o; o += (size_t)H * 352;      // 32 x 352
    (void)ws_size; (void)n_in;

    const int TB = 256;

    // --- zero scratch for OOB GEMM lanes ---
    fill_f32<<<cdiv(ZBUF_N, TB), TB, 0, stream>>>(zbuf, 0.0f, ZBUF_N);

    // --- weight prep (transpose + zero-pad K) ---
    transpose_pad<<<cdiv(H * XPAD, TB), TB, 0, stream>>>(W[0], W0t, F, H, XPAD);
    transpose_pad<<<cdiv(H * H, TB), TB, 0, stream>>>(W[1], W1t, H, H, H);
    transpose_pad<<<cdiv(H * H, TB), TB, 0, stream>>>(W[2], W2t, H, H, H);
    transpose_pad<<<cdiv(1 * H, TB), TB, 0, stream>>>(W[3], W3t, H, 1, H);
    pad_rows<<<cdiv(16 * PSTRIDE, TB), TB, 0, stream>>>(conv1_w, C1t, 16, D_TOT, D_TOT, PSTRIDE);
    transpose_pad<<<cdiv(H * 352, TB), TB, 0, stream>>>(mlp_w1, M1t, 352, H, 352);
    // --- pad x to N x 24 ---
    pad_rows<<<cdiv((long long)N * XPAD, TB), TB, 0, stream>>>(x, xpad, N, F, F, XPAD);

    // --- degree / normalization ---
    fill_f32<<<cdiv(N, TB), TB, 0, stream>>>(dis, 1.0f, N);   // self loop
    deg_scatter<<<cdiv(E, TB), TB, 0, stream>>>(dst, dis, E);
    deg_to_dis<<<cdiv(N, TB), TB, 0, stream>>>(dis, N);

    // --- 4 GCN layers ---
    for (int l = 0; l < 4; ++l) {
        const int dout  = (l == 3) ? 1 : H;
        const int shift = (l == 3) ? 0 : 5;
        const float* A   = (l == 0) ? xpad : (feat + (size_t)(l - 1) * H);
        const int    lda = (l == 0) ? XPAD : FSTRIDE;
        const int    Kd  = (l == 0) ? XPAD : H;
        const float* Bt  = (l == 0) ? W0t : (l == 1) ? W1t : (l == 2) ? W2t : W3t;
        const int    ldb = (l == 0) ? XPAD : H;

        launch_gemm(A, lda, Bt, ldb, nullptr, zbuf, hw, dout, N, Kd, dout, 0, stream);
        gcn_init<<<cdiv((long long)N << shift, TB), TB, 0, stream>>>(hw, dis, bW[l], agg, N, shift);
        gcn_scatter<<<cdiv((long long)E << shift, TB), TB, 0, stream>>>(hw, dis, src, dst, agg, E, shift);
        gcn_tanh<<<cdiv((long long)N << shift, TB), TB, 0, stream>>>(agg, feat, N, shift, l * H);
    }

    // --- SortPooling (top-K per graph) ---
    sortpool_kernel<<<G, 128, 0, stream>>>(feat, batch, pooled, N);

    // --- conv1 (stride==kernel==97) == (G*30, 104) @ (104, 16) + b, ReLU ---
    launch_gemm(pooled, PSTRIDE, C1t, PSTRIDE, conv1_b, zbuf, y1, 16,
                G * KTOP, PSTRIDE, 16, 1, stream);

    // --- maxpool k=2 ---
    maxpool_kernel<<<cdiv((long long)G * 16 * 15, TB), TB, 0, stream>>>(y1, y1p, G);

    // --- conv2 + relu + flatten ---
    conv2_kernel<<<cdiv((long long)G * 32 * 11, TB), TB, 0, stream>>>(y1p, conv2_w, conv2_b, y2, G);

    // --- MLP1: (G,352)@(352,32)+b, ReLU ---
    launch_gemm(y2, 352, M1t, 352, mlp_b1, zbuf, m1, 32, G, 352, 32, 1, stream);

    // --- MLP2: (G,32)@(32,1)+b ---
    mlp2_kernel<<<cdiv(G, TB), TB, 0, stream>>>(m1, mlp_w2, mlp_b2, (float*)d_out, G);
}